// SelfAttention_19593640805109
// MI455X (gfx1250) — compile-verified
//
#include <hip/hip_runtime.h>

// Problem constants (match reference)
#define B_ 16
#define N_ 512
#define F_ 256
#define H_ 8
#define E_ 16
#define D_ 32
#define BIG_NEG -1000000000.0f

typedef float v2f __attribute__((ext_vector_type(2)));
typedef float v8f __attribute__((ext_vector_type(8)));
typedef int   v4i __attribute__((ext_vector_type(4)));

// ---------------------------------------------------------------------------
// CDNA5 async global->LDS copy (ASYNCcnt path), with safe fallback.
// Probe-confirmed: param0 is 'AS(1) int4*' (clang prints AS1 as __device__).
// ---------------------------------------------------------------------------
#if defined(__has_builtin)
#  if __has_builtin(__builtin_amdgcn_global_load_async_to_lds_b128)
#    define HAVE_ASYNC_COPY 1
#  endif
#endif
#ifndef HAVE_ASYNC_COPY
#  define HAVE_ASYNC_COPY 0
#endif

typedef __attribute__((address_space(1))) v4i gbl_v4i;
typedef __attribute__((address_space(3))) v4i lds_v4i;

__device__ __forceinline__ void copy16_to_lds(const float* gsrc, float* ldst) {
#if HAVE_ASYNC_COPY
    __builtin_amdgcn_global_load_async_to_lds_b128(
        (gbl_v4i*)gsrc, (lds_v4i*)ldst, /*offset=*/0, /*cpol=*/0);
#else
    *(float4*)ldst = *(const float4*)gsrc;
#endif
}

__device__ __forceinline__ void wait_async_copies() {
#if HAVE_ASYNC_COPY
#  if __has_builtin(__builtin_amdgcn_s_wait_asynccnt)
    __builtin_amdgcn_s_wait_asynccnt(0);
#  else
    asm volatile("s_wait_asynccnt 0" ::: "memory");
#  endif
#endif
}

// D = A(16x4, f32) x B(4x16, f32) + C(16x16, f32)
// Per-lane: A/B = v2f (lanes 0-15: K=0,1 ; lanes 16-31: K=2,3), C/D = v8f
__device__ __forceinline__ v8f wmma_f32_k4(v2f a, v2f b, v8f c) {
    return __builtin_amdgcn_wmma_f32_16x16x4_f32(
        /*neg_a=*/false, a, /*neg_b=*/false, b,
        /*c_mod=*/(short)0, c, /*reuse_a=*/false, /*reuse_b=*/false);
}

// ---------------------------------------------------------------------------
// Kernel 1 & 3: row-major GEMM  C[M,NOUT] = A[M,KDIM] @ W[KDIM,NOUT] + bias
// One wave computes a 16x64 tile (4 accumulators). Optional row-mask multiply.
// ---------------------------------------------------------------------------
template <int KDIM, int NOUT, bool APPLY_MASK>
__global__ __launch_bounds__(256) void gemm_kernel(
    const float* __restrict__ A, const float* __restrict__ W,
    const float* __restrict__ bias, const int* __restrict__ mask,
    float* __restrict__ C) {
    const int lane   = threadIdx.x & 31;
    const int wid    = threadIdx.x >> 5;
    const int laneLo = lane & 15;
    const int laneHi = lane >> 4;
    const int ctiles = NOUT / 64;

    const int gw   = blockIdx.x * 8 + wid;
    const int rt   = gw / ctiles;
    const int ct   = gw % ctiles;
    const int row0 = rt * 16;
    const int col0 = ct * 64;

    v8f acc[4];
    acc[0] = v8f{}; acc[1] = v8f{}; acc[2] = v8f{}; acc[3] = v8f{};

    const int m = row0 + laneLo;
    #pragma unroll 4
    for (int k0 = 0; k0 < KDIM; k0 += 4) {
        const int kk = k0 + 2 * laneHi;
        v2f a = *(const v2f*)(A + (size_t)m * KDIM + kk);
        const float* w0 = W + (size_t)kk * NOUT + col0 + laneLo;
        const float* w1 = w0 + NOUT;
        v2f b0 = {w0[0],  w1[0]};
        v2f b1 = {w0[16], w1[16]};
        v2f b2 = {w0[32], w1[32]};
        v2f b3 = {w0[48], w1[48]};
        acc[0] = wmma_f32_k4(a, b0, acc[0]);
        acc[1] = wmma_f32_k4(a, b1, acc[1]);
        acc[2] = wmma_f32_k4(a, b2, acc[2]);
        acc[3] = wmma_f32_k4(a, b3, acc[3]);
    }

    #pragma unroll
    for (int t = 0; t < 4; ++t) {
        const int c  = col0 + t * 16 + laneLo;
        const float bv = bias[c];
        #pragma unroll
        for (int r = 0; r < 8; ++r) {
            const int row = row0 + r + 8 * laneHi;
            float o = acc[t][r] + bv;
            if constexpr (APPLY_MASK) { o *= (float)mask[row]; }
            C[(size_t)row * NOUT + c] = o;
        }
    }
}

// ---------------------------------------------------------------------------
// Kernel 2: fused edge-biased flash attention.
// Grid: B * (N/16) blocks; 256 threads = 8 waves; wave h handles head h.
// edges read exactly once (shared across all heads in the block).
// qkv layout: [B,N,768] (q: cols 0..255, k: 256..511, v: 512..767).
// ---------------------------------------------------------------------------
__global__ __launch_bounds__(256) void attn_kernel(
    const float* __restrict__ qkv, const float* __restrict__ edges,
    const float* __restrict__ Wb, const float* __restrict__ bb,
    const int* __restrict__ mask, float* __restrict__ attnout) {
    __shared__ float sK[16][260];       // K tile, padded rows (bank-conflict-free)
    __shared__ float sV[16][260];       // V tile
    __shared__ float sBias[16 * 16 * 9];// bias tile [i][j][h], h-stride 9
    __shared__ float sP[8 * 16 * 20];   // per-head P tile, row stride 20
    __shared__ float sWb[E_ * H_];
    __shared__ float sbb[H_];

    const int lane   = threadIdx.x & 31;
    const int h      = threadIdx.x >> 5;   // wave id == head id
    const int laneLo = lane & 15;
    const int laneHi = lane >> 4;
    const int b      = blockIdx.x >> 5;    // 32 i-tiles per batch
    const int i0     = (blockIdx.x & 31) * 16;

    if (threadIdx.x < E_ * H_) sWb[threadIdx.x] = Wb[threadIdx.x];
    if (threadIdx.x < H_)      sbb[threadIdx.x] = bb[threadIdx.x];

    // Q fragments for this head (scale 1/sqrt(D) folded in)
    const float scale = 0.17677669529663687f;  // 1/sqrt(32)
    v2f qa[8];
    {
        const int m = i0 + laneLo;
        const float* qrow = qkv + ((size_t)(b * N_) + m) * 768 + h * D_;
        #pragma unroll
        for (int s = 0; s < 8; ++s) {
            const int kk = 4 * s + 2 * laneHi;
            v2f q = *(const v2f*)(qrow + kk);
            qa[s][0] = q[0] * scale;
            qa[s][1] = q[1] * scale;
        }
    }

    v8f o0 = v8f{}, o1 = v8f{};
    float rowM[8], rowL[8];
    #pragma unroll
    for (int r = 0; r < 8; ++r) { rowM[r] = -1e30f; rowL[r] = 0.0f; }

    __syncthreads();

    for (int jt = 0; jt < 32; ++jt) {
        const int j0 = jt * 16;

        // ---- Phase 1: async K/V tile copies (ASYNCcnt path) + bias tile ----
        #pragma unroll
        for (int t = 0; t < 4; ++t) {
            const int idx = t * 256 + threadIdx.x;   // 0..1023 float4 chunks
            const int j   = idx >> 6;
            const int c4  = (idx & 63) * 4;
            const float* base = qkv + ((size_t)(b * N_) + j0 + j) * 768;
            copy16_to_lds(base + 256 + c4, &sK[j][c4]);
            copy16_to_lds(base + 512 + c4, &sV[j][c4]);
        }
        {
            const int ii = threadIdx.x >> 4;
            const int jj = threadIdx.x & 15;
            const int gi = i0 + ii;
            const int gj = j0 + jj;
            const float* erow = edges + ((size_t)(b * N_ + gi)) * N_ * E_;
            // prefetch next j-tile's edge vector for this (i,j) slot (HBM stream)
            if (jt + 1 < 32) {
                __builtin_prefetch(erow + (size_t)(j0 + 16 + jj) * E_, 0, 3);
            }
            float acc[H_];
            #pragma unroll
            for (int hh = 0; hh < H_; ++hh) acc[hh] = 0.0f;
            if (gi != gj) {  // self-edges are zeroed before Wb projection
                const float* ep = erow + (size_t)gj * E_;
                #pragma unroll
                for (int e = 0; e < E_; ++e) {
                    const float v = ep[e];
                    #pragma unroll
                    for (int hh = 0; hh < H_; ++hh) acc[hh] += v * sWb[e * H_ + hh];
                }
            }
            float* bp = &sBias[(ii * 16 + jj) * 9];
            #pragma unroll
            for (int hh = 0; hh < H_; ++hh) bp[hh] = acc[hh] + sbb[hh];
        }
        wait_async_copies();   // this wave's async tile bytes are in LDS
        __syncthreads();       // ... and visible to all waves

        // ---- Phase 2: S = Q K^T (per head), bias+mask, online softmax ----
        v8f s = v8f{};
        #pragma unroll
        for (int ss = 0; ss < 8; ++ss) {
            const int kk = 4 * ss + 2 * laneHi;
            v2f bfr = *(const v2f*)(&sK[laneLo][h * D_ + kk]);
            s = wmma_f32_k4(qa[ss], bfr, s);
        }
        const float mterm =
            (1.0f - (float)mask[b * N_ + j0 + laneLo]) * BIG_NEG;
        #pragma unroll
        for (int r = 0; r < 8; ++r) {
            const int row = r + 8 * laneHi;
            float x = s[r] + sBias[(row * 16 + laneLo) * 9 + h] + mterm;
            // row max over 16 lanes (halves stay separate under wave32)
            float mx = x;
            mx = fmaxf(mx, __shfl_xor(mx, 1, 32));
            mx = fmaxf(mx, __shfl_xor(mx, 2, 32));
            mx = fmaxf(mx, __shfl_xor(mx, 4, 32));
            mx = fmaxf(mx, __shfl_xor(mx, 8, 32));
            const float mnew = fmaxf(rowM[r], mx);
            const float corr = __expf(rowM[r] - mnew);
            const float p    = __expf(x - mnew);
            float ps = p;
            ps += __shfl_xor(ps, 1, 32);
            ps += __shfl_xor(ps, 2, 32);
            ps += __shfl_xor(ps, 4, 32);
            ps += __shfl_xor(ps, 8, 32);
            rowL[r] = rowL[r] * corr + ps;
            rowM[r] = mnew;
            o0[r] *= corr;
            o1[r] *= corr;
            sP[h * 320 + row * 20 + laneLo] = p;  // C-layout -> LDS
        }
        __syncthreads();

        // ---- Phase 3: O += P V (P re-read in A-fragment layout) ----
        #pragma unroll
        for (int ss = 0; ss < 4; ++ss) {
            const int kk = 4 * ss + 2 * laneHi;
            v2f pa = *(const v2f*)(&sP[h * 320 + laneLo * 20 + kk]);
            v2f vb0 = {sV[kk][h * D_ + laneLo], sV[kk + 1][h * D_ + laneLo]};
            v2f vb1 = {sV[kk][h * D_ + 16 + laneLo],
                       sV[kk + 1][h * D_ + 16 + laneLo]};
            o0 = wmma_f32_k4(pa, vb0, o0);
            o1 = wmma_f32_k4(pa, vb1, o1);
        }
        __syncthreads();  // protect tiles/sP before next iteration overwrites
    }

    // ---- Epilogue: normalize and store [B,N,F] (head-interleaved) ----
    #pragma unroll
    for (int r = 0; r < 8; ++r) {
        const int row = r + 8 * laneHi;
        const float inv = 1.0f / rowL[r];
        float* orow = attnout + ((size_t)(b * N_) + i0 + row) * F_ + h * D_;
        orow[laneLo]      = o0[r] * inv;
        orow[16 + laneLo] = o1[r] * inv;
    }
}

// ---------------------------------------------------------------------------
// Host launcher
// ---------------------------------------------------------------------------
extern "C" void kernel_launch(void* const* d_in, const int* in_sizes, int n_in,
                              void* d_out, int out_size, void* d_ws,
                              size_t ws_size, hipStream_t stream) {
    (void)in_sizes; (void)n_in; (void)out_size; (void)ws_size;
    const float* nodes = (const float*)d_in[0];
    const float* edges = (const float*)d_in[1];
    const float* Wqkv  = (const float*)d_in[2];
    const float* bqkv  = (const float*)d_in[3];
    const float* Wb    = (const float*)d_in[4];
    const float* bb    = (const float*)d_in[5];
    const float* Wo    = (const float*)d_in[6];
    const float* bo    = (const float*)d_in[7];
    const int*   nmask = (const int*)d_in[8];
    float* out = (float*)d_out;

    float* qkv     = (float*)d_ws;                       // [B,N,768]  24 MB
    float* attnout = qkv + (size_t)B_ * N_ * 3 * F_;     // [B,N,256]   8 MB

    // QKV projection: 8192x256 @ 256x768 -> 512*12 wave-tiles / 8 = 768 blocks
    gemm_kernel<256, 768, false>
        <<<768, 256, 0, stream>>>(nodes, Wqkv, bqkv, nullptr, qkv);

    // Fused edge-biased attention: B * N/16 = 512 blocks
    attn_kernel<<<512, 256, 0, stream>>>(qkv, edges, Wb, bb, nmask, attnout);

    // Output projection + mask: 8192x256 @ 256x256 -> 512*4/8 = 256 blocks
    gemm_kernel<256, 256, true>
        <<<256, 256, 0, stream>>>(attnout, Wo, bo, nmask, out);
}